// FiberBundleEmbedding_65635690217750
// MI455X (gfx1250) — compile-verified
//
#include <hip/hip_runtime.h>

// FiberBundleEmbedding for MI455X (gfx1250, wave32).
//  out = [ base gather | polar(fiber gather) | connection gather | generators ]
// Polar factor via 4 Newton-Schulz iterations using V_WMMA_F32_16X16X4_F32.

#define VOCAB   50257
#define D_BASE  512
#define D_FIBER 32
#define LIE_RANK 8
#define BATCH   8
#define SEQ     4096
#define NTOK    (BATCH * SEQ)           // 32768 matrices
#define NS_ITERS 4
#define WAVES_PER_BLOCK 4

typedef __attribute__((ext_vector_type(2))) float v2f;
typedef __attribute__((ext_vector_type(8))) float v8f;

__device__ __forceinline__ v8f wmma4(v2f a, v2f b, v8f c) {
  // D = A(16x4) * B(4x16) + C(16x16), f32, wave32
  return __builtin_amdgcn_wmma_f32_16x16x4_f32(
      /*neg_a=*/false, a, /*neg_b=*/false, b,
      /*c_mod=*/(short)0, c, /*reuse_a=*/false, /*reuse_b=*/false);
}

// ---------------- bandwidth kernels (coalesced float4 gathers) --------------

__global__ void fb_gather_base(const int* __restrict__ tokens,
                               const float4* __restrict__ basew,
                               float4* __restrict__ out) {
  int t = blockIdx.x * blockDim.x + threadIdx.x;   // NTOK * 128 float4s
  int tokIdx = t >> 7;                             // 512 floats = 128 float4 / row
  int col    = t & 127;
  int tok    = tokens[tokIdx];
  out[t] = basew[(size_t)tok * 128 + col];
}

__global__ void fb_gather_conn(const int* __restrict__ tokens,
                               const float4* __restrict__ connw,
                               float4* __restrict__ out) {
  int t = blockIdx.x * blockDim.x + threadIdx.x;   // NTOK * 2 float4s
  int tok = tokens[t >> 1];
  out[t] = connw[(size_t)tok * 2 + (t & 1)];
}

__global__ void fb_copy_gen(const float4* __restrict__ g,
                            float4* __restrict__ out) {
  int t = blockIdx.x * blockDim.x + threadIdx.x;   // 2048 float4s
  out[t] = g[t];
}

// ---------------- polar decomposition kernel --------------------------------
// One wave32 per 32x32 matrix. Matrix lives in this wave's private LDS slice;
// ping-pong buffers + Z buffer (3 * 4KB). No cross-wave sharing -> no barriers
// (LDS ops are in-order within a wave; compiler inserts s_wait_dscnt).

__global__ void fb_polar_fiber(const int* __restrict__ tokens,
                               const float* __restrict__ fiberw,
                               float* __restrict__ outf) {
  __shared__ __align__(16) float lds[WAVES_PER_BLOCK][3][D_FIBER * D_FIBER];

  const int lane = threadIdx.x & 31;
  const int wave = threadIdx.x >> 5;
  const int mi   = blockIdx.x * WAVES_PER_BLOCK + wave;
  const int tok  = tokens[mi];

  float* Ya = &lds[wave][0][0];
  float* Yb = &lds[wave][1][0];
  float* Z  = &lds[wave][2][0];

  // ---- load 32x32 tile (coalesced float4) into LDS ----
  {
    const float4* src = (const float4*)(fiberw + (size_t)tok * 1024);
    float4*       dst = (float4*)Ya;
#pragma unroll
    for (int i = 0; i < 8; ++i) dst[i * 32 + lane] = src[i * 32 + lane];
  }

  const int l  = lane & 15;   // N (B/C) or M (A) index
  const int kh = lane >> 4;   // K half-select per ISA f32 operand layout

  float* Ycur  = Ya;
  float* Ynext = Yb;

#pragma unroll
  for (int it = 0; it < NS_ITERS; ++it) {
    // ---- Z = Y^T Y : four 16x16 blocks, 8 K-chunks of 4 each ----
#pragma unroll
    for (int bi = 0; bi < 2; ++bi) {
#pragma unroll
      for (int bj = 0; bj < 2; ++bj) {
        v8f acc = {0.f, 0.f, 0.f, 0.f, 0.f, 0.f, 0.f, 0.f};
#pragma unroll
        for (int kc = 0; kc < 8; ++kc) {
          int k0 = kc * 4 + kh * 2;
          v2f a, b;
          // A = (Y^T) block: A(m,k) = Y[k][16*bi + m]  (transpose is free)
          a.x = Ycur[(k0    ) * 32 + bi * 16 + l];
          a.y = Ycur[(k0 + 1) * 32 + bi * 16 + l];
          // B = Y block:     B(k,n) = Y[k][16*bj + n]
          b.x = Ycur[(k0    ) * 32 + bj * 16 + l];
          b.y = Ycur[(k0 + 1) * 32 + bj * 16 + l];
          acc = wmma4(a, b, acc);
        }
        // C-layout scatter: element (M = r + 8*kh, N = l)
#pragma unroll
        for (int r = 0; r < 8; ++r)
          Z[(bi * 16 + r + kh * 8) * 32 + bj * 16 + l] = acc[r];
      }
    }

    // ---- Ynext = 1.5*Y - 0.5*(Y @ Z) ----
#pragma unroll
    for (int bi = 0; bi < 2; ++bi) {
#pragma unroll
      for (int bj = 0; bj < 2; ++bj) {
        v8f acc = {0.f, 0.f, 0.f, 0.f, 0.f, 0.f, 0.f, 0.f};
#pragma unroll
        for (int kc = 0; kc < 8; ++kc) {
          int k0 = kc * 4 + kh * 2;
          v2f a, b;
          // A(m,k) = Y[16*bi + m][k] : two consecutive floats -> ds_load_b64
          a = *(const v2f*)&Ycur[(bi * 16 + l) * 32 + k0];
          // B(k,n) = Z[k][16*bj + n]
          b.x = Z[(k0    ) * 32 + bj * 16 + l];
          b.y = Z[(k0 + 1) * 32 + bj * 16 + l];
          acc = wmma4(a, b, acc);
        }
#pragma unroll
        for (int r = 0; r < 8; ++r) {
          int gm = bi * 16 + r + kh * 8;
          int gn = bj * 16 + l;
          float y = Ycur[gm * 32 + gn];
          Ynext[gm * 32 + gn] = 1.5f * y - 0.5f * acc[r];
        }
      }
    }
    float* t2 = Ycur; Ycur = Ynext; Ynext = t2;
  }

  // ---- store result (coalesced float4) ----
  {
    float4*       dst = (float4*)(outf + (size_t)mi * 1024);
    const float4* src = (const float4*)Ycur;
#pragma unroll
    for (int i = 0; i < 8; ++i) dst[i * 32 + lane] = src[i * 32 + lane];
  }
}

// ---------------- launcher ---------------------------------------------------

extern "C" void kernel_launch(void* const* d_in, const int* in_sizes, int n_in,
                              void* d_out, int out_size, void* d_ws, size_t ws_size,
                              hipStream_t stream) {
  const int*   tokens = (const int*)d_in[0];
  const float* basew  = (const float*)d_in[1];
  const float* fiberw = (const float*)d_in[2];
  const float* connw  = (const float*)d_in[3];
  const float* gen    = (const float*)d_in[4];

  float* out       = (float*)d_out;
  float* out_base  = out;
  float* out_fiber = out_base + (size_t)NTOK * D_BASE;
  float* out_conn  = out_fiber + (size_t)NTOK * D_FIBER * D_FIBER;
  float* out_gen   = out_conn + (size_t)NTOK * LIE_RANK;

  // base gather: NTOK*128 float4s
  fb_gather_base<<<(NTOK * 128) / 256, 256, 0, stream>>>(
      tokens, (const float4*)basew, (float4*)out_base);

  // connection gather: NTOK*2 float4s
  fb_gather_conn<<<(NTOK * 2) / 256, 256, 0, stream>>>(
      tokens, (const float4*)connw, (float4*)out_conn);

  // generators passthrough: 8192 floats = 2048 float4s
  fb_copy_gen<<<2048 / 256, 256, 0, stream>>>(
      (const float4*)gen, (float4*)out_gen);

  // polar: one wave per matrix, 4 waves (128 threads) per block
  fb_polar_fiber<<<NTOK / WAVES_PER_BLOCK, WAVES_PER_BLOCK * 32, 0, stream>>>(
      tokens, fiberw, out_fiber);
}